// Gate_81277961110149
// MI455X (gfx1250) — compile-verified
//
#include <hip/hip_runtime.h>

#define TOKENS       131072
#define HIDDEN       2048
#define NUM_EXPERTS  64
#define TOPK         8
#define ROUTER_SCALE 2.5f

#define TOK_PER_WAVE  64                         // 4 A-tiles of 16 tokens
#define WAVES_PER_BLK 8
#define TOK_PER_BLOCK (TOK_PER_WAVE * WAVES_PER_BLK)   // 512
#define EPI_ROWS      128                        // tokens staged per epilogue phase
#define ROWSTRIDE     65                         // padded: conflict-free column scans

typedef float v2f __attribute__((ext_vector_type(2)));
typedef float v8f __attribute__((ext_vector_type(8)));

__global__ __launch_bounds__(256, 2)
void moe_gate_wmma_kernel(const float* __restrict__ x,
                          const float* __restrict__ w,
                          const float* __restrict__ bias,
                          float* __restrict__ w_out,
                          int*   __restrict__ idx_out)
{
    __shared__ float logits_s[EPI_ROWS * ROWSTRIDE];   // 33.3 KB
    __shared__ float bias_s[NUM_EXPERTS];

    const int tid  = threadIdx.x;
    const int wave = tid >> 5;          // wave32
    const int lane = tid & 31;
    const int half = lane >> 4;         // 0: K={0,1}, 1: K={2,3}
    const int lid  = lane & 15;         // M (A) / N (B) index within 16-wide tile

    if (tid < NUM_EXPERTS) bias_s[tid] = bias[tid];

    const int tok_base = blockIdx.x * TOK_PER_BLOCK + wave * TOK_PER_WAVE;

    // A tile m: x[tok_base + 16*m + lid][k + 2*half .. +1]  (16x4 f32 A layout)
    const float* aptr[4];
    #pragma unroll
    for (int m = 0; m < 4; ++m)
        aptr[m] = x + (size_t)(tok_base + 16 * m + lid) * HIDDEN + 2 * half;

    // B tile t: w[16*t + lid][k + 2*half .. +1]             (4x16 f32 B layout)
    const float* bptr[4];
    #pragma unroll
    for (int t = 0; t < 4; ++t)
        bptr[t] = w + (size_t)(16 * t + lid) * HIDDEN + 2 * half;

    v8f acc[4][4] = {};   // [A tile m][B tile t] -> 16 x (16x16 f32) = 128 VGPRs

    #pragma unroll 4
    for (int k = 0; k < HIDDEN; k += 4) {
        v2f a[4], b[4];
        #pragma unroll
        for (int m = 0; m < 4; ++m) a[m] = *(const v2f*)(aptr[m] + k);
        #pragma unroll
        for (int t = 0; t < 4; ++t) b[t] = *(const v2f*)(bptr[t] + k);

        #pragma unroll
        for (int m = 0; m < 4; ++m)
            #pragma unroll
            for (int t = 0; t < 4; ++t)
                // (neg_a, A, neg_b, B, c_mod, C, reuse_a, reuse_b)
                acc[m][t] = __builtin_amdgcn_wmma_f32_16x16x4_f32(
                    false, a[m], false, b[t], (short)0, acc[m][t], false, false);
    }

    // Phased epilogue: 128 tokens (one A-tile per wave) staged per phase.
    // C/D layout: VGPR r, lanes 0-15 -> (M=r, N=lid); lanes 16-31 -> (M=8+r, N=lid)
    #pragma unroll
    for (int p = 0; p < 4; ++p) {
        #pragma unroll
        for (int r = 0; r < 8; ++r) {
            const int trow = wave * 16 + r + 8 * half;
            logits_s[trow * ROWSTRIDE + lid +  0] = acc[p][0][r];
            logits_s[trow * ROWSTRIDE + lid + 16] = acc[p][1][r];
            logits_s[trow * ROWSTRIDE + lid + 32] = acc[p][2][r];
            logits_s[trow * ROWSTRIDE + lid + 48] = acc[p][3][r];
        }
        __syncthreads();

        if (tid < EPI_ROWS) {
            float* row = &logits_s[tid * ROWSTRIDE];

            float mx = row[0];
            #pragma unroll
            for (int e = 1; e < NUM_EXPERTS; ++e) mx = fmaxf(mx, row[e]);

            float sum = 0.0f;
            #pragma unroll
            for (int e = 0; e < NUM_EXPERTS; ++e) {
                float v = expf(row[e] - mx);
                row[e] = v;
                sum += v;
            }
            const float inv = 1.0f / sum;
            #pragma unroll
            for (int e = 0; e < NUM_EXPERTS; ++e) row[e] *= inv;  // unbiased probs

            // buffer row tid = (wave tid>>4, tile-row tid&15) of phase p
            const int tok = blockIdx.x * TOK_PER_BLOCK
                          + (tid >> 4) * TOK_PER_WAVE + p * 16 + (tid & 15);

            unsigned long long sel = 0ull;
            #pragma unroll
            for (int j = 0; j < TOPK; ++j) {
                float best = -__builtin_inff();
                int   be   = 0;
                for (int e = 0; e < NUM_EXPERTS; ++e) {
                    const float v = row[e] + bias_s[e];            // biased selection
                    const bool taken = (sel >> e) & 1ull;
                    if (!taken && v > best) { best = v; be = e; }  // top_k tie-break
                }
                sel |= (1ull << be);
                w_out[(size_t)tok * TOPK + j]   = row[be] * ROUTER_SCALE;
                idx_out[(size_t)tok * TOPK + j] = be;
            }
        }
        __syncthreads();
    }
}

extern "C" void kernel_launch(void* const* d_in, const int* in_sizes, int n_in,
                              void* d_out, int out_size, void* d_ws, size_t ws_size,
                              hipStream_t stream)
{
    const float* x    = (const float*)d_in[0];   // [131072, 2048]
    const float* w    = (const float*)d_in[1];   // [64, 2048]
    const float* bias = (const float*)d_in[2];   // [64]

    float* w_out   = (float*)d_out;                                  // [131072, 8] f32
    int*   idx_out = (int*)((float*)d_out + (size_t)TOKENS * TOPK);  // [131072, 8] i32

    dim3 grid(TOKENS / TOK_PER_BLOCK);   // 256 blocks
    dim3 block(256);                     // 8 wave32 waves
    moe_gate_wmma_kernel<<<grid, block, 0, stream>>>(x, w, bias, w_out, idx_out);
}